// CIF_Model_67869073212220
// MI455X (gfx1250) — compile-verified
//
#include <hip/hip_runtime.h>

#define B_ 32
#define T_ 2000
#define H_ 512
#define L_ 256          // max_label_len
#define THRESH 0.95f

typedef __attribute__((ext_vector_type(2))) float v2f;
typedef __attribute__((ext_vector_type(8))) float v8f;

// ---------------------------------------------------------------------------
// Phase 1: scalar integrate-and-fire scan, one lane per batch row (wave32).
// Packs per-timestep metadata into one 16B record for a single b128 load in
// phase 2:  meta[t] = { w1, w2, k1 (bit-cast), 0 }
//   w1 -> weight of h[t] into frame k1   (cur if fire, alpha otherwise)
//   w2 -> weight of h[t] into frame k1+1 (remainder; nonzero only on fire)
// ---------------------------------------------------------------------------
__global__ __launch_bounds__(32) void cif_scan(
    const float* __restrict__ alphas, float4* __restrict__ meta,
    int* __restrict__ fireT, int* __restrict__ nfires)
{
    int b = threadIdx.x;
    if (b >= B_) return;
    const float* ar = alphas + (size_t)b * T_;
    float4*      mr = meta   + (size_t)b * T_;
    float integ = 0.0f;
    int   k     = 0;
    for (int t = 0; t < T_; ++t) {
        float a        = ar[t];
        float integNew = integ + a;
        bool  fire     = integNew > THRESH;
        float cur      = fire ? (1.0f - integ) : a;
        float rem      = fire ? (a - cur)      : 0.0f;
        mr[t] = make_float4(cur, rem, __int_as_float(k), 0.0f);
        if (fire) {
            if (k < L_) fireT[b * L_ + k] = t;
            ++k;
        }
        integ = fire ? (integNew - 1.0f) : integNew;
    }
    nfires[b] = k;
}

// ---------------------------------------------------------------------------
// Phase 2: banded SpMM  out[b] (256x512) = W[b] (256x2000) * hidden[b] (2000x512)
// via V_WMMA_F32_16X16X4_F32. One wave per block = (batch b, 16-frame tile m0,
// 64-column tile n0). K-loop covers only the contiguous fire segment feeding
// frames [m0, m0+16), 4 timesteps per chunk.
//
// Main loop invariant (no clamping needed): for t in [tStart, tEnd],
// k1[t] <= kEnd < limit, so the w1 predicate is pure band equality and the
// w2 predicate folds to (k1+1 == mrow) && (mrow < limit)  [hoisted].
// Reads past tEnd only happen in the single tail chunk, where clamped
// addresses + validity selects handle them.
// ---------------------------------------------------------------------------
__global__ __launch_bounds__(32) void cif_spmm_wmma(
    const float* __restrict__ hidden, const float4* __restrict__ meta,
    const int* __restrict__ fireT, const int* __restrict__ nfires,
    float* __restrict__ out)
{
    const int b    = blockIdx.x;
    const int m0   = blockIdx.y << 4;   // frame-tile base (0..240)
    const int n0   = blockIdx.z << 6;   // H-column base (0..448)
    const int lane = threadIdx.x;
    const int half = lane >> 4;         // 0: K={0,1}, 1: K={2,3}
    const int lrow = lane & 15;
    const int mrow = m0 + lrow;         // this lane's A-matrix row

    v8f acc0 = {}; v8f acc1 = {}; v8f acc2 = {}; v8f acc3 = {};

    int limit = nfires[b];
    if (limit > L_) limit = L_;
    const bool mValid = (mrow < limit); // w2 target validity, hoisted

    // Segment feeding frames [m0, m0+16): uniform per block -> EXEC all-ones
    // at every WMMA.
    int tStart = 0, tEnd = -1;
    if (m0 < limit) {
        tStart   = (m0 == 0) ? 0 : fireT[b * L_ + (m0 - 1)];
        int kEnd = m0 + 15; if (kEnd > limit - 1) kEnd = limit - 1;
        tEnd     = fireT[b * L_ + kEnd];
    }

    const int len   = tEnd - tStart + 1;          // <= 0 when tile empty
    const int nFull = (len > 0) ? (len >> 2) : 0; // full 4-step chunks
    const int nTail = (len > 0) ? (len & 3)  : 0;

    // Per-lane streaming pointers; advance by pure increments in the loop.
    const float4* mp = meta + (size_t)b * T_ + tStart + (half << 1);
    const float*  hp = hidden + ((size_t)b * T_ + tStart + (half << 1)) * H_
                              + n0 + lrow;

    for (int c = 0; c < nFull; ++c) {
        float4 ma = mp[0];                 // t = t0 + 2*half     (b128)
        float4 mb = mp[1];                 // t = t0 + 2*half + 1 (b128)
        int kA = __float_as_int(ma.z);
        int kB = __float_as_int(mb.z);

        float ax = (kA     == mrow          ) ? ma.x : 0.0f;
        ax      += (kA + 1 == mrow && mValid) ? ma.y : 0.0f;
        float ay = (kB     == mrow          ) ? mb.x : 0.0f;
        ay      += (kB + 1 == mrow && mValid) ? mb.y : 0.0f;
        v2f A = {ax, ay};

        // 8 immediate-offset b32 loads off one base (two hidden rows).
        v2f B0 = {hp[0],  hp[H_ +  0]};
        v2f B1 = {hp[16], hp[H_ + 16]};
        v2f B2 = {hp[32], hp[H_ + 32]};
        v2f B3 = {hp[48], hp[H_ + 48]};

        acc0 = __builtin_amdgcn_wmma_f32_16x16x4_f32(false, A, false, B0, (short)0, acc0, false, false);
        acc1 = __builtin_amdgcn_wmma_f32_16x16x4_f32(false, A, false, B1, (short)0, acc1, false, false);
        acc2 = __builtin_amdgcn_wmma_f32_16x16x4_f32(false, A, false, B2, (short)0, acc2, false, false);
        acc3 = __builtin_amdgcn_wmma_f32_16x16x4_f32(false, A, false, B3, (short)0, acc3, false, false);

        mp += 4;
        hp += 4 * H_;
    }

    if (nTail) {
        const int t0 = tStart + (nFull << 2);
        const int tA = t0 + (half << 1);
        const int tB = tA + 1;
        const int tAc = (tA <= tEnd) ? tA : tEnd;   // clamp into segment
        const int tBc = (tB <= tEnd) ? tB : tEnd;
        const bool vA = (tA <= tEnd);
        const bool vB = (tB <= tEnd);

        const float4* mbase = meta + (size_t)b * T_;
        float4 ma = mbase[tAc];
        float4 mb = mbase[tBc];
        int kA = __float_as_int(ma.z);
        int kB = __float_as_int(mb.z);

        float ax = (vA && kA     == mrow          ) ? ma.x : 0.0f;
        ax      += (vA && kA + 1 == mrow && mValid) ? ma.y : 0.0f;
        float ay = (vB && kB     == mrow          ) ? mb.x : 0.0f;
        ay      += (vB && kB + 1 == mrow && mValid) ? mb.y : 0.0f;
        v2f A = {ax, ay};

        const float* hbase = hidden + (size_t)b * T_ * H_ + n0 + lrow;
        const float* rA = hbase + (size_t)tAc * H_;
        const float* rB = hbase + (size_t)tBc * H_;
        v2f B0 = {rA[0],  rB[0]};
        v2f B1 = {rA[16], rB[16]};
        v2f B2 = {rA[32], rB[32]};
        v2f B3 = {rA[48], rB[48]};

        acc0 = __builtin_amdgcn_wmma_f32_16x16x4_f32(false, A, false, B0, (short)0, acc0, false, false);
        acc1 = __builtin_amdgcn_wmma_f32_16x16x4_f32(false, A, false, B1, (short)0, acc1, false, false);
        acc2 = __builtin_amdgcn_wmma_f32_16x16x4_f32(false, A, false, B2, (short)0, acc2, false, false);
        acc3 = __builtin_amdgcn_wmma_f32_16x16x4_f32(false, A, false, B3, (short)0, acc3, false, false);
    }

    // Store 16x64 tile; every output element written exactly once across the
    // grid, frames beyond nfires come out as exact zeros (zero weights).
    float* ob = out + (size_t)b * L_ * H_;
    #pragma unroll
    for (int i = 0; i < 8; ++i) {
        int row    = m0 + i + (half << 3);
        size_t off = (size_t)row * H_ + n0 + lrow;
        ob[off]      = acc0[i];
        ob[off + 16] = acc1[i];
        ob[off + 32] = acc2[i];
        ob[off + 48] = acc3[i];
    }
}

extern "C" void kernel_launch(void* const* d_in, const int* in_sizes, int n_in,
                              void* d_out, int out_size, void* d_ws, size_t ws_size,
                              hipStream_t stream) {
    const float* hidden = (const float*)d_in[0];  // [B, T, H] fp32
    const float* alphas = (const float*)d_in[1];  // [B, T]    fp32
    float* out = (float*)d_out;                   // [B, L, H] fp32

    // Workspace layout (~1.06 MB total); meta first for 16B alignment.
    float4* meta   = (float4*)d_ws;                   // B*T  (16B records)
    int*    fireT  = (int*)(meta + (size_t)B_ * T_);  // B*L
    int*    nfires = fireT + (size_t)B_ * L_;         // B

    cif_scan<<<dim3(1), dim3(32), 0, stream>>>(alphas, meta, fireT, nfires);

    dim3 grid(B_, L_ / 16, H_ / 64);
    cif_spmm_wmma<<<grid, dim3(32), 0, stream>>>(hidden, meta, fireT,
                                                 nfires, out);
}